// SparseWeightedAdjacency_86655260164600
// MI455X (gfx1250) — compile-verified
//
#include <hip/hip_runtime.h>
#include <math.h>

#define T_LEN 8
#define N_PED 1024
#define DMODEL 64
#define HEADS 4
#define HDIM 16            // DMODEL / HEADS
#define RAD2DEG 57.29577951308232f

typedef float v2f __attribute__((ext_vector_type(2)));
typedef float v4f __attribute__((ext_vector_type(4)));
typedef float v8f __attribute__((ext_vector_type(8)));

// ---------------------------------------------------------------------------
// Kernel 1: spatial projections.  row = t*N + n, x = graph[t,n,1:3] (2 feats)
// e = x @ ew^T + eb ; q = e @ qw^T + qb ; k = e @ kw^T + kb
// blockDim = 64 (one row per block; thread c owns channel c)
// ---------------------------------------------------------------------------
__global__ void proj_spatial(const float* __restrict__ graph,
                             const float* __restrict__ ew, const float* __restrict__ eb,
                             const float* __restrict__ qw, const float* __restrict__ qb,
                             const float* __restrict__ kw, const float* __restrict__ kb,
                             float* __restrict__ qout, float* __restrict__ kout) {
    __shared__ float e[DMODEL];
    const int row = blockIdx.x;          // 0 .. T*N-1
    const int c   = threadIdx.x;         // 0 .. 63
    const float x0 = graph[row * 3 + 1];
    const float x1 = graph[row * 3 + 2];
    e[c] = fmaf(x0, ew[c * 2 + 0], fmaf(x1, ew[c * 2 + 1], eb[c]));
    __syncthreads();
    float q = qb[c];
    float k = kb[c];
#pragma unroll 8
    for (int d = 0; d < DMODEL; ++d) {
        const float ed = e[d];
        q = fmaf(ed, qw[c * DMODEL + d], q);
        k = fmaf(ed, kw[c * DMODEL + d], k);
    }
    qout[row * DMODEL + c] = q;
    kout[row * DMODEL + c] = k;
}

// ---------------------------------------------------------------------------
// Spatial mask term, computed analytically (identity_spatial never read).
// ---------------------------------------------------------------------------
__device__ __forceinline__ float spa_mask_val(int i, int j, float sgx, float sgy,
                                              float na, float oix, float oiy,
                                              const float* __restrict__ obs, int t) {
    const float dx = obs[(j * 2 + 0) * T_LEN + t] - oix;
    const float dy = obs[(j * 2 + 1) * T_LEN + t] - oiy;
    const float dist = sqrtf(fmaf(dx, dx, dy * dy));
    const float nb = fmaxf(dist, 1e-8f);
    float sim = fmaf(sgx, dx, sgy * dy) / (na * nb);
    const float diag = (i == j) ? 1.0f : 0.0f;
    sim += diag;                                      // + eye(N) before arccos
    const float deg = acosf(fminf(fmaxf(sim, -1.0f), 1.0f)) * RAD2DEG;
    const float sd = (dist < 5.0f) ? 1.0f : 0.0f;
    return ((deg * sd) < 10.0f ? 1.0f : 0.0f) + diag; // + identity_spatial
}

// ---------------------------------------------------------------------------
// Kernel 2: fused spatial attention.
// grid = T*HEADS*(N/16) blocks, 256 threads (8 wave32).
// Per block: 16 output rows x 1024 cols for one (t,h).
//   Phase A: QK^T via V_WMMA_F32_16X16X4_F32 (K=16 -> 4 chained WMMAs/tile),
//            A/B fragments loaded as b64 (the lane's two K-slots are adjacent).
//   Phase B: softmax -> *mask (on-the-fly) -> softmax; all LDS sweeps and the
//            final (bandwidth-critical) global write are float4 / b128.
// dyn LDS: att[16*1024] + red[256] + rstat[16]
// ---------------------------------------------------------------------------
__global__ void spatial_attn(const float* __restrict__ qbuf,
                             const float* __restrict__ kbuf,
                             const float* __restrict__ graph,
                             const float* __restrict__ obs,
                             float* __restrict__ out) {
    extern __shared__ float smem[];
    float* att   = smem;                 // 16 * 1024
    float* red   = smem + 16 * 1024;     // 256
    float* rstat = red + 256;            // 16

    const int blk   = blockIdx.x;
    const int iBase = (blk & 63) * 16;          // row-tile
    const int h     = (blk >> 6) & 3;
    const int t     = blk >> 8;

    const int tid  = threadIdx.x;
    const int wave = tid >> 5;
    const int lane = tid & 31;
    const int laneM = lane & 15;
    const bool hi   = lane >= 16;
    const int kOff  = hi ? 2 : 0;    // lane's K slot pair: {0,1} lo-half, {2,3} hi-half

    // ---- Phase A: WMMA QK^T ------------------------------------------------
    // A-matrix (q tile) fragments: identical for all 8 j-tiles of this wave.
    const float* qrow = qbuf + ((size_t)(t * N_PED + iBase + laneM)) * DMODEL + h * HDIM + kOff;
    const v2f a0 = *(const v2f*)(qrow + 0);
    const v2f a1 = *(const v2f*)(qrow + 4);
    const v2f a2 = *(const v2f*)(qrow + 8);
    const v2f a3 = *(const v2f*)(qrow + 12);

    for (int s = 0; s < 8; ++s) {
        const int jBase = (wave * 8 + s) * 16;
        const float* krow = kbuf + ((size_t)(t * N_PED + jBase + laneM)) * DMODEL + h * HDIM + kOff;
        const v2f b0 = *(const v2f*)(krow + 0);
        const v2f b1 = *(const v2f*)(krow + 4);
        const v2f b2 = *(const v2f*)(krow + 8);
        const v2f b3 = *(const v2f*)(krow + 12);

        v8f acc = {};
        acc = __builtin_amdgcn_wmma_f32_16x16x4_f32(false, a0, false, b0, (short)0, acc, false, false);
        acc = __builtin_amdgcn_wmma_f32_16x16x4_f32(false, a1, false, b1, (short)0, acc, false, false);
        acc = __builtin_amdgcn_wmma_f32_16x16x4_f32(false, a2, false, b2, (short)0, acc, false, false);
        acc = __builtin_amdgcn_wmma_f32_16x16x4_f32(false, a3, false, b3, (short)0, acc, false, false);

        // D layout: VGPR r -> row r (lanes 0-15) / row r+8 (lanes 16-31), col = lane&15
        const int rbase = hi ? 8 : 0;
#pragma unroll
        for (int r = 0; r < 8; ++r)
            att[(rbase + r) * N_PED + jBase + laneM] = acc[r] * 0.125f;   // /sqrt(64)
    }
    __syncthreads();

    // ---- Phase B: softmax -> mask -> softmax -> store ----------------------
    // 16 threads per row; each thread owns 16 interleaved float4 chunks:
    // chunk c = it*16 + seg  ->  columns [4c, 4c+3].
    const int row = tid >> 4;            // 0..15
    const int seg = tid & 15;            // 0..15
    const int gi  = iBase + row;
    float* arow = att + row * N_PED;
    v4f* arow4  = (v4f*)arow;

    // pass 1: row max of att
    float m = -3.4e38f;
#pragma unroll 4
    for (int it = 0; it < 16; ++it) {
        const v4f x = arow4[it * 16 + seg];
        m = fmaxf(m, fmaxf(fmaxf(x.x, x.y), fmaxf(x.z, x.w)));
    }
    red[tid] = m;
    __syncthreads();
    if (seg == 0) {
        float mm = red[row * 16];
        for (int c = 1; c < 16; ++c) mm = fmaxf(mm, red[row * 16 + c]);
        rstat[row] = mm;
    }
    __syncthreads();
    const float m1 = rstat[row];

    // pass 2: row sum of exp
    float ssum = 0.0f;
#pragma unroll 4
    for (int it = 0; it < 16; ++it) {
        const v4f x = arow4[it * 16 + seg];
        ssum += __expf(x.x - m1) + __expf(x.y - m1) + __expf(x.z - m1) + __expf(x.w - m1);
    }
    red[tid] = ssum;
    __syncthreads();
    if (seg == 0) {
        float sm = 0.0f;
        for (int c = 0; c < 16; ++c) sm += red[row * 16 + c];
        rstat[row] = sm;
    }
    __syncthreads();
    const float inv_s1 = 1.0f / rstat[row];
    __syncthreads();

    // pass 3: v = dense * mask (mask generated on the fly), track max(v)
    const float sgx = graph[(t * N_PED + gi) * 3 + 1];
    const float sgy = graph[(t * N_PED + gi) * 3 + 2];
    const float na  = fmaxf(sqrtf(fmaf(sgx, sgx, sgy * sgy)), 1e-8f);
    const float oix = obs[(gi * 2 + 0) * T_LEN + t];
    const float oiy = obs[(gi * 2 + 1) * T_LEN + t];
    float m2 = -3.4e38f;
    for (int it = 0; it < 16; ++it) {
        const int ch = it * 16 + seg;
        v4f x = arow4[ch];
        const int j0 = ch * 4;
#pragma unroll
        for (int u = 0; u < 4; ++u) {
            const float dense = __expf(x[u] - m1) * inv_s1;
            const float v = dense * spa_mask_val(gi, j0 + u, sgx, sgy, na, oix, oiy, obs, t);
            x[u] = v;
            m2 = fmaxf(m2, v);
        }
        arow4[ch] = x;
    }
    red[tid] = m2;
    __syncthreads();
    if (seg == 0) {
        float mm = red[row * 16];
        for (int c = 1; c < 16; ++c) mm = fmaxf(mm, red[row * 16 + c]);
        rstat[row] = mm;
    }
    __syncthreads();
    m2 = rstat[row];

    // pass 4: sum2
    float s2 = 0.0f;
#pragma unroll 4
    for (int it = 0; it < 16; ++it) {
        const v4f x = arow4[it * 16 + seg];
        s2 += __expf(x.x - m2) + __expf(x.y - m2) + __expf(x.z - m2) + __expf(x.w - m2);
    }
    red[tid] = s2;
    __syncthreads();
    if (seg == 0) {
        float sm = 0.0f;
        for (int c = 0; c < 16; ++c) sm += red[row * 16 + c];
        rstat[row] = sm;
    }
    __syncthreads();
    const float inv_s2 = 1.0f / rstat[row];

    // pass 5: final write, b128 fully-coalesced (only touch of the 134MB output)
    v4f* orow4 = (v4f*)(out + (((size_t)(t * HEADS + h) * N_PED) + gi) * N_PED);
#pragma unroll 4
    for (int it = 0; it < 16; ++it) {
        const int ch = it * 16 + seg;
        const v4f x = arow4[ch];
        v4f o;
        o.x = __expf(x.x - m2) * inv_s2;
        o.y = __expf(x.y - m2) * inv_s2;
        o.z = __expf(x.z - m2) * inv_s2;
        o.w = __expf(x.w - m2) * inv_s2;
        orow4[ch] = o;
    }
}

// ---------------------------------------------------------------------------
// Kernel 3: temporal branch, fully fused per pedestrian n (8x8 attention).
// blockDim = 64, grid = N.
// ---------------------------------------------------------------------------
__global__ void temporal_kernel(const float* __restrict__ graph,
                                const float* __restrict__ ew, const float* __restrict__ eb,
                                const float* __restrict__ qw, const float* __restrict__ qb,
                                const float* __restrict__ kw, const float* __restrict__ kb,
                                float* __restrict__ out) {
    __shared__ float e[T_LEN * DMODEL];
    __shared__ float q[T_LEN * DMODEL];
    __shared__ float k[T_LEN * DMODEL];
    __shared__ float attb[HEADS * T_LEN * T_LEN];
    __shared__ float mask[T_LEN * T_LEN];

    const int n = blockIdx.x;
    const int c = threadIdx.x;   // 0..63

    // e[t][c]
#pragma unroll
    for (int t = 0; t < T_LEN; ++t) {
        const float x0 = graph[(t * N_PED + n) * 3 + 0];
        const float x1 = graph[(t * N_PED + n) * 3 + 1];
        const float x2 = graph[(t * N_PED + n) * 3 + 2];
        e[t * DMODEL + c] = fmaf(x0, ew[c * 3 + 0],
                            fmaf(x1, ew[c * 3 + 1],
                            fmaf(x2, ew[c * 3 + 2], eb[c])));
    }
    __syncthreads();

#pragma unroll
    for (int t = 0; t < T_LEN; ++t) {
        float qv = qb[c];
        float kv = kb[c];
        for (int d = 0; d < DMODEL; ++d) {
            const float ed = e[t * DMODEL + d];
            qv = fmaf(ed, qw[c * DMODEL + d], qv);
            kv = fmaf(ed, kw[c * DMODEL + d], kv);
        }
        q[t * DMODEL + c] = qv;
        k[t * DMODEL + c] = kv;
    }
    __syncthreads();

    // att[h][i][j], 256 entries -> 4 per thread
#pragma unroll
    for (int rep = 0; rep < 4; ++rep) {
        const int idx = rep * 64 + c;
        const int h = idx >> 6;
        const int i = (idx >> 3) & 7;
        const int j = idx & 7;
        float s = 0.0f;
        for (int d = 0; d < HDIM; ++d)
            s = fmaf(q[i * DMODEL + h * HDIM + d], k[j * DMODEL + h * HDIM + d], s);
        attb[idx] = s * 0.125f;
    }

    // temporal mask (identity_temporal never read; borders are 0 + eye)
    {
        const int i = c >> 3, j = c & 7;
        float mv = (i == j) ? 1.0f : 0.0f;
        if (i >= 1 && j >= 1) {
            const float vix = graph[(i * N_PED + n) * 3 + 1];
            const float viy = graph[(i * N_PED + n) * 3 + 2];
            const float vjx = graph[(j * N_PED + n) * 3 + 1];
            const float vjy = graph[(j * N_PED + n) * 3 + 2];
            const float na = fmaxf(sqrtf(fmaf(vix, vix, viy * viy)), 1e-8f);
            const float nb = fmaxf(sqrtf(fmaf(vjx, vjx, vjy * vjy)), 1e-8f);
            const float sim = fmaf(vix, vjx, viy * vjy) / (na * nb);
            const float deg = acosf(fminf(fmaxf(sim, -1.0f), 1.0f)) * RAD2DEG;
            mv += (deg > 10.0f) ? 1.0f : 0.0f;
        }
        mask[c] = mv;
    }
    __syncthreads();

    // double softmax, one row of 8 per thread (32 rows used)
    if (c < HEADS * T_LEN) {
        const int h = c >> 3, i = c & 7;
        float r[T_LEN];
        float m1 = -3.4e38f;
#pragma unroll
        for (int j = 0; j < T_LEN; ++j) { r[j] = attb[h * 64 + i * 8 + j]; m1 = fmaxf(m1, r[j]); }
        float s1 = 0.0f;
#pragma unroll
        for (int j = 0; j < T_LEN; ++j) { r[j] = __expf(r[j] - m1); s1 += r[j]; }
        const float inv1 = 1.0f / s1;
        float m2 = -3.4e38f;
#pragma unroll
        for (int j = 0; j < T_LEN; ++j) { r[j] = r[j] * inv1 * mask[i * 8 + j]; m2 = fmaxf(m2, r[j]); }
        float s2 = 0.0f;
#pragma unroll
        for (int j = 0; j < T_LEN; ++j) { r[j] = __expf(r[j] - m2); s2 += r[j]; }
        const float inv2 = 1.0f / s2;
        float* orow = out + (((size_t)(n * HEADS + h) * T_LEN) + i) * T_LEN;
#pragma unroll
        for (int j = 0; j < T_LEN; ++j) orow[j] = r[j] * inv2;
    }
}

// ---------------------------------------------------------------------------
extern "C" void kernel_launch(void* const* d_in, const int* in_sizes, int n_in,
                              void* d_out, int out_size, void* d_ws, size_t ws_size,
                              hipStream_t stream) {
    (void)in_sizes; (void)n_in; (void)out_size; (void)ws_size;
    const float* graph     = (const float*)d_in[0];
    // d_in[1] identity_spatial, d_in[2] identity_temporal: regenerated analytically
    const float* obs       = (const float*)d_in[3];
    const float* spa_emb_w = (const float*)d_in[4];
    const float* spa_emb_b = (const float*)d_in[5];
    const float* spa_q_w   = (const float*)d_in[6];
    const float* spa_q_b   = (const float*)d_in[7];
    const float* spa_k_w   = (const float*)d_in[8];
    const float* spa_k_b   = (const float*)d_in[9];
    const float* tem_emb_w = (const float*)d_in[10];
    const float* tem_emb_b = (const float*)d_in[11];
    const float* tem_q_w   = (const float*)d_in[12];
    const float* tem_q_b   = (const float*)d_in[13];
    const float* tem_k_w   = (const float*)d_in[14];
    const float* tem_k_b   = (const float*)d_in[15];

    float* spa_q = (float*)d_ws;                       // T*N*64 floats (2MB)
    float* spa_k = spa_q + (size_t)T_LEN * N_PED * DMODEL;

    float* outS = (float*)d_out;                                   // (T,4,N,N)
    float* outT = outS + (size_t)T_LEN * HEADS * N_PED * N_PED;    // (N,4,T,T)

    proj_spatial<<<T_LEN * N_PED, 64, 0, stream>>>(
        graph, spa_emb_w, spa_emb_b, spa_q_w, spa_q_b, spa_k_w, spa_k_b, spa_q, spa_k);

    const size_t smem = (16 * 1024 + 256 + 16) * sizeof(float);    // ~65KB
    spatial_attn<<<T_LEN * HEADS * (N_PED / 16), 256, smem, stream>>>(
        spa_q, spa_k, graph, obs, outS);

    temporal_kernel<<<N_PED, 64, 0, stream>>>(
        graph, tem_emb_w, tem_emb_b, tem_q_w, tem_q_b, tem_k_w, tem_k_b, outT);
}